// NFPPooling_63986422776070
// MI455X (gfx1250) — compile-verified
//
#include <hip/hip_runtime.h>

// Problem constants (from reference)
#define BATCH 8
#define CHN   128
#define HH    224
#define WW    224
#define HO    222
#define WO    222

// Tiling
#define TH      16            // pixel rows per block
#define TWPIX   64            // pixel cols per block (4 per thread in x)
#define G       8             // channels staged per LDS buffer
#define NSTAGES (CHN / G)     // 16
#define LROWS   17            // tile rows staged (pixel rows + 1)
#define LCOLSP  68            // staged cols: gcols [tj0-2 .. tj0+65] (even-aligned origin)
#define LSTRIDE 68            // LDS row stride in words
#define CH_PAIRS (LROWS * (LCOLSP / 2))  // 578 b64 pairs per channel
#define CH_LDS_W (LROWS * LSTRIDE)       // 1156 words of LDS per channel
#define SLOTS   3                        // b64 async issues per thread per channel (3*256 >= 578)
#define STAGE_ISSUES (G * SLOTS)         // 24 -> max ASYNCcnt in flight = 48 < 63

__global__ __launch_bounds__(256) void nfp_l1_stencil(const float* __restrict__ x,
                                                      float* __restrict__ out)
{
    __shared__ float smem[2 * G * CH_LDS_W];   // double buffer, 2 x 36992 B

    const int tx  = threadIdx.x;               // 0..15 (4-col strips)
    const int ty  = threadIdx.y;               // 0..15 (rows)
    const int tid = ty * 16 + tx;
    const int tj0 = blockIdx.x * TWPIX;        // pixel-col base (0,64,128,192)
    const int ti0 = blockIdx.y * TH;           // pixel-row base (0..208)
    const int b   = blockIdx.z;

    // ---- Hoisted async-copy address pattern (identical for every channel) ----
    // Tile covers global rows [ti0 .. ti0+16], cols [tj0-2 .. tj0+65], 8B-aligned pairs.
    uint32_t gOffB[SLOTS], ldsOffB[SLOTS];
#pragma unroll
    for (int k = 0; k < SLOTS; ++k) {
        int p = tid + k * 256;
        if (p > CH_PAIRS - 1) p = CH_PAIRS - 1;          // duplicate tail (benign)
        int row = p / 34;
        int cp  = p - row * 34;
        int grow = ti0 + row;       if (grow > HH - 1) grow = HH - 1;
        int gc   = tj0 - 2 + cp * 2;                     // even
        if (gc < 0) gc = 0;                              // stays even
        if (gc > WW - 2) gc = WW - 2;                    // stays even, pair in-bounds
        gOffB[k]   = (uint32_t)((grow * WW + gc) * 4);   // 8B aligned
        ldsOffB[k] = (uint32_t)((row * LSTRIDE + cp * 2) * 4);
    }

    const uint32_t smemBase = (uint32_t)(size_t)(&smem[0]);  // low 32 bits = LDS byte offset
    const uint32_t bufBytes = (uint32_t)(G * CH_LDS_W * 4);
    const float* xb = x + (size_t)b * CHN * (HH * WW);

    // ---- Issue stage 0 (24 async b64 copies per thread) ----
    {
#pragma unroll
        for (int ci = 0; ci < G; ++ci) {
            const float* src = xb + ci * (HH * WW);
#pragma unroll
            for (int k = 0; k < SLOTS; ++k) {
                uint32_t l = smemBase + (uint32_t)(ci * CH_LDS_W * 4) + ldsOffB[k];
                asm volatile("global_load_async_to_lds_b64 %0, %1, %2 th:TH_LOAD_NT"
                             :: "v"(l), "v"(gOffB[k]), "s"(src) : "memory");
            }
        }
    }

    // Accumulators: 4 canonical directions x 4 pixels
    float a01[4]  = {0.f, 0.f, 0.f, 0.f};   // d = ( 0, +1)
    float a1m1[4] = {0.f, 0.f, 0.f, 0.f};   // d = (+1, -1)
    float a10[4]  = {0.f, 0.f, 0.f, 0.f};   // d = (+1,  0)
    float a11[4]  = {0.f, 0.f, 0.f, 0.f};   // d = (+1, +1)

    for (int s = 0; s < NSTAGES; ++s) {
        // Prefetch next stage into the other buffer, then wait for current stage.
        if (s + 1 < NSTAGES) {
            const float* src0 = xb + (size_t)(s + 1) * G * (HH * WW);
            uint32_t base = smemBase + (((s + 1) & 1) ? bufBytes : 0u);
#pragma unroll
            for (int ci = 0; ci < G; ++ci) {
                const float* src = src0 + ci * (HH * WW);
#pragma unroll
                for (int k = 0; k < SLOTS; ++k) {
                    uint32_t l = base + (uint32_t)(ci * CH_LDS_W * 4) + ldsOffB[k];
                    asm volatile("global_load_async_to_lds_b64 %0, %1, %2 th:TH_LOAD_NT"
                                 :: "v"(l), "v"(gOffB[k]), "s"(src) : "memory");
                }
            }
            // 24 new ops in flight; ASYNCcnt <= 24 => previous stage's 24 complete (in-order).
            asm volatile("s_wait_asynccnt 0x18" ::: "memory");
        } else {
            asm volatile("s_wait_asynccnt 0x0" ::: "memory");
        }
        __syncthreads();

        const float* buf = smem + ((s & 1) ? G * CH_LDS_W : 0);
#pragma unroll
        for (int ci = 0; ci < G; ++ci) {
            const float* tb = buf + ci * CH_LDS_W + ty * LSTRIDE + tx * 4;
            // Two aligned ds_load_b128 per row: local cols [tx*4 .. tx*4+7]
            float4 v0 = *(const float4*)(tb);
            float4 v1 = *(const float4*)(tb + 4);
            float4 u0 = *(const float4*)(tb + LSTRIDE);
            float4 u1 = *(const float4*)(tb + LSTRIDE + 4);
            float r0[8] = {v0.x, v0.y, v0.z, v0.w, v1.x, v1.y, v1.z, v1.w};
            float r1[8] = {u0.x, u0.y, u0.z, u0.w, u1.x, u1.y, u1.z, u1.w};
#pragma unroll
            for (int k = 0; k < 4; ++k) {
                float a = r0[k + 2];                 // x[i, j]   (j = tj0 + tx*4 + k)
                a01[k]  += fabsf(a - r0[k + 3]);     // vs x[i,   j+1]
                a1m1[k] += fabsf(a - r1[k + 1]);     // vs x[i+1, j-1]
                a10[k]  += fabsf(a - r1[k + 2]);     // vs x[i+1, j  ]
                a11[k]  += fabsf(a - r1[k + 3]);     // vs x[i+1, j+1]
            }
        }
        __syncthreads();
    }

    // ---- Epilogue: each direction sum feeds two symmetric output planes ----
    // offsets o: 0:(0,0) 1:(0,1) 2:(0,2) 3:(1,0) 4:(1,2) 5:(2,0) 6:(2,1) 7:(2,2)
    // canonical d=(0,1)->planes 4 & 3, (1,-1)->5 & 2, (1,0)->6 & 1, (1,1)->7 & 0
    const int i  = ti0 + ty;
    const int jb = tj0 + tx * 4;
    const size_t plane = (size_t)HO * WO;
    float* ob = out + (size_t)b * 8 * plane;

#pragma unroll
    for (int k = 0; k < 4; ++k) {
        const int j = jb + k;
        // Positive planes 4..7 at (i-1, j-1)
        if (i >= 1 && i <= HO && j >= 1 && j <= WO) {
            size_t t = (size_t)(i - 1) * WO + (size_t)(j - 1);
            __builtin_nontemporal_store(-a01[k],  &ob[4 * plane + t]);
            __builtin_nontemporal_store(-a1m1[k], &ob[5 * plane + t]);
            __builtin_nontemporal_store(-a10[k],  &ob[6 * plane + t]);
            __builtin_nontemporal_store(-a11[k],  &ob[7 * plane + t]);
        }
        // Mirrored planes 3..0 at (i+di-1, j+dj-1)
        if (i >= 1 && i <= HO && j < WO)
            __builtin_nontemporal_store(-a01[k],  &ob[3 * plane + (size_t)(i - 1) * WO + (size_t)j]);
        if (i < HO && j >= 2)
            __builtin_nontemporal_store(-a1m1[k], &ob[2 * plane + (size_t)i * WO + (size_t)(j - 2)]);
        if (i < HO && j >= 1 && j <= WO)
            __builtin_nontemporal_store(-a10[k],  &ob[1 * plane + (size_t)i * WO + (size_t)(j - 1)]);
        if (i < HO && j < WO)
            __builtin_nontemporal_store(-a11[k],  &ob[0 * plane + (size_t)i * WO + (size_t)j]);
    }
}

extern "C" void kernel_launch(void* const* d_in, const int* in_sizes, int n_in,
                              void* d_out, int out_size, void* d_ws, size_t ws_size,
                              hipStream_t stream) {
    (void)in_sizes; (void)n_in; (void)d_ws; (void)ws_size; (void)out_size;
    const float* x = (const float*)d_in[0];
    float* out = (float*)d_out;
    dim3 grid(4, 14, BATCH);     // 4 col-tiles (covers 256, clamped), 14 row-tiles, 8 batches
    dim3 block(16, 16, 1);       // 256 threads = 8 wave32
    nfp_l1_stencil<<<grid, block, 0, stream>>>(x, out);
}